// MMDGLM_38637525795131
// MI455X (gfx1250) — compile-verified
//
#include <hip/hip_runtime.h>
#include <hip/hip_bf16.h>

typedef float v2f __attribute__((ext_vector_type(2)));
typedef float v8f __attribute__((ext_vector_type(8)));

#define T_TOT    1024
#define N_FR_    2048
#define N_D_     2048
#define N_PAR    20
#define KTILES   (N_FR_ / 16)
#define GBLK     256

// ---------------------------------------------------------------------------
// Kernel 1: per-(t,k) log-likelihood terms, reduced over a t-tile.
// One wave (32 threads) handles 16 consecutive k for a range of t.
// u[t,k] = X[t,k,:] . theta computed with v_wmma_f32_16x16x4_f32:
//   A = X-rows pre-scaled by theta (per-lane), B = all-ones, C chained over
//   five K=4 chunks  =>  D[i][*] = dot(X_row_i, theta), replicated over N.
// ---------------------------------------------------------------------------
__global__ __launch_bounds__(32)
void mmdglm_ll_kernel(const float* __restrict__ X,
                      const int*   __restrict__ mask,
                      const float* __restrict__ bptr,
                      const float* __restrict__ kappa,
                      const float* __restrict__ dtp,
                      float*       __restrict__ ws_ll,
                      int NT)
{
    const int lane = threadIdx.x;              // wave32: 0..31
    const int kt   = blockIdx.x % KTILES;
    const int tt   = blockIdx.x / KTILES;
    const int k0   = kt * 16;
    const int tspan = T_TOT / NT;
    const int t0    = tt * tspan;

    const float dt = dtp[0];

    // theta = concat(b, kappa)
    float th[N_PAR];
    th[0] = bptr[0];
#pragma unroll
    for (int i = 1; i < N_PAR; ++i) th[i] = kappa[i - 1];

    const int  mrow  = lane & 15;              // A-matrix M row held by this lane
    const bool hi    = lane >= 16;             // half-wave selects K slots {2,3}
    const int  khalf = hi ? 2 : 0;

    // per-lane theta pair matching the A K-slots this lane supplies
    v2f tv[5];
#pragma unroll
    for (int c = 0; c < 5; ++c) {
        tv[c].x = hi ? th[4 * c + 2] : th[4 * c + 0];
        tv[c].y = hi ? th[4 * c + 3] : th[4 * c + 1];
    }

    const v2f ones = {1.0f, 1.0f};             // B matrix: layout-proof all-ones

    // lanes 0..7 own rows 0..7 (D VGPRs 0..7, lanes 0-15 duplicated);
    // lanes 16..23 own rows 8..15 (D VGPRs 0..7, lanes 16-31 duplicated)
    const bool active = ((lane & 15) < 8);
    const int  r      = lane & 7;
    const int  ksel   = k0 + r + (hi ? 8 : 0);

    float facc = 0.0f;

    for (int t = t0; t < t0 + tspan; ++t) {
        const float* rowp = X + ((size_t)t * N_FR_ + (size_t)(k0 + mrow)) * N_PAR + khalf;
        v8f acc = {0.f, 0.f, 0.f, 0.f, 0.f, 0.f, 0.f, 0.f};
#pragma unroll
        for (int c = 0; c < 5; ++c) {
            v2f a = *(const v2f*)(rowp + 4 * c);   // 8B load, 8B aligned
            a *= tv[c];                            // pre-scale by theta chunk
            acc = __builtin_amdgcn_wmma_f32_16x16x4_f32(
                false, a, false, ones, (short)0, acc, false, false);
        }
        if (active) {
            float u = (r & 4) ? ((r & 2) ? ((r & 1) ? acc[7] : acc[6])
                                         : ((r & 1) ? acc[5] : acc[4]))
                              : ((r & 2) ? ((r & 1) ? acc[3] : acc[2])
                                         : ((r & 1) ? acc[1] : acc[0]));
            float rr = expf(u);
            float x  = dt * rr;
            int   mk = mask[(size_t)t * N_FR_ + ksel];
            facc += mk ? log1pf(-expf(-x)) : -x;
        }
    }
    if (active) ws_ll[(size_t)tt * N_FR_ + ksel] = facc;
}

// ---------------------------------------------------------------------------
// Kernel 2: strict upper-triangle sum of gramian_d_d -> 256 block partials.
// ---------------------------------------------------------------------------
__global__ __launch_bounds__(256)
void triu_sum_kernel(const float* __restrict__ G, float* __restrict__ out)
{
    __shared__ float red[256];
    float s = 0.0f;
    for (int i = blockIdx.x; i < N_D_; i += GBLK) {
        const float* row = G + (size_t)i * N_D_;
        for (int j = i + 1 + (int)threadIdx.x; j < N_D_; j += 256)
            s += row[j];
    }
    red[threadIdx.x] = s;
    __syncthreads();
    for (int w = 128; w > 0; w >>= 1) {
        if ((int)threadIdx.x < w) red[threadIdx.x] += red[threadIdx.x + w];
        __syncthreads();
    }
    if (threadIdx.x == 0) out[blockIdx.x] = red[0];
}

// ---------------------------------------------------------------------------
// Kernel 3: reduce ll partials + phi sums + gram partials, emit (loss, mmd).
// ---------------------------------------------------------------------------
__global__ __launch_bounds__(256)
void finalize_kernel(const float* __restrict__ ws_ll, int NT,
                     const float* __restrict__ ws_gram,
                     const float* __restrict__ phi_fr,
                     const float* __restrict__ phi_d,
                     const int*   __restrict__ lamp,
                     float*       __restrict__ out)
{
    __shared__ float red[256];
    float Sl = 0.f, Q = 0.f, S = 0.f, P = 0.f, Sd = 0.f, Gs = 0.f;

    for (int k = threadIdx.x; k < N_FR_; k += 256) {
        float ll = 0.f;
        for (int tk = 0; tk < NT; ++tk) ll += ws_ll[(size_t)tk * N_FR_ + k];
        float p = phi_fr[k];
        Sl += ll * p;
        Q  += ll * p * p;
        S  += p;
        P  += p * p;
    }
    for (int i = threadIdx.x; i < N_D_; i += 256) Sd += phi_d[i];
    Gs = ws_gram[threadIdx.x];   // GBLK == 256 partials

    float vals[6] = {Sl, Q, S, P, Sd, Gs};
    float res[6];
#pragma unroll
    for (int v = 0; v < 6; ++v) {
        red[threadIdx.x] = vals[v];
        __syncthreads();
        for (int w = 128; w > 0; w >>= 1) {
            if ((int)threadIdx.x < w) red[threadIdx.x] += red[threadIdx.x + w];
            __syncthreads();
        }
        res[v] = red[0];
        __syncthreads();
    }

    if (threadIdx.x == 0) {
        const float lam        = (float)lamp[0];
        const float denom_half = (float)N_FR_ * (float)(N_FR_ - 1) * 0.5f; // == N_D version
        const float crossN     = (float)N_D_ * (float)N_FR_;
        // sum_{i<j}(ll_i+ll_j)phi_i phi_j = Sl*S - Q ; mean cross = Sd*Sl/crossN
        float mmd_grad = (res[0] * res[2] - res[1]) / denom_half
                       - 2.0f * res[4] * res[0] / crossN;
        float mmd = res[5] / denom_half
                  + (res[2] * res[2] - res[3]) / (2.0f * denom_half)
                  - 2.0f * res[4] * res[2] / crossN;
        out[0] = lam * mmd_grad;
        out[1] = mmd;
    }
}

extern "C" void kernel_launch(void* const* d_in, const int* in_sizes, int n_in,
                              void* d_out, int out_size, void* d_ws, size_t ws_size,
                              hipStream_t stream)
{
    const float* X      = (const float*)d_in[0];   // [T, N_FR, 20]
    const int*   mask   = (const int*)  d_in[1];   // [T, N_FR]
    const float* phi_fr = (const float*)d_in[2];   // [N_FR]
    const float* phi_d  = (const float*)d_in[3];   // [N_D]
    const float* gram   = (const float*)d_in[4];   // [N_D, N_D]
    const float* bptr   = (const float*)d_in[5];   // [1]
    const float* kappa  = (const float*)d_in[6];   // [19]
    const float* dtp    = (const float*)d_in[7];   // [1]
    const int*   lamp   = (const int*)  d_in[8];   // scalar 1

    // t-tiles: as many as workspace permits (power of two dividing 1024)
    int NT = 32;
    while (NT > 1 && (size_t)(NT * N_FR_ + GBLK) * sizeof(float) > ws_size) NT >>= 1;

    float* ws_ll   = (float*)d_ws;                  // [NT * N_FR]
    float* ws_gram = ws_ll + (size_t)NT * N_FR_;    // [GBLK]

    mmdglm_ll_kernel<<<KTILES * NT, 32, 0, stream>>>(X, mask, bptr, kappa, dtp, ws_ll, NT);
    triu_sum_kernel<<<GBLK, 256, 0, stream>>>(gram, ws_gram);
    finalize_kernel<<<1, 256, 0, stream>>>(ws_ll, NT, ws_gram, phi_fr, phi_d, lamp, (float*)d_out);
}